// RelConv_67413806678363
// MI455X (gfx1250) — compile-verified
//
#include <hip/hip_runtime.h>

typedef __attribute__((ext_vector_type(2))) float v2f;
typedef __attribute__((ext_vector_type(8))) float v8f;

#define D_FEAT 64

// ---------------------------------------------------------------------------
// GEMM kernel: one wave computes one 16-row tile of BOTH
//   h    = x @ W1^T            (written to workspace)
//   out  = x @ Wr^T + br       (written to d_out)
// using V_WMMA_F32_16X16X4_F32. A (16x4 f32) and B (4x16 f32) fragments are
// 2 VGPRs each; lane L holds K = 2*(L/16) + vgpr, so both are float2 loads.
// Tail handling is wave-uniform so the hot path has straight-line stores and
// EXEC stays all-1s across every WMMA.
// ---------------------------------------------------------------------------
__global__ __launch_bounds__(256) void relconv_gemm_wmma(
    const float* __restrict__ x,
    const float* __restrict__ W1,
    const float* __restrict__ Wr,
    const float* __restrict__ br,
    float* __restrict__ h,
    float* __restrict__ out,
    int n_nodes, int mtiles) {
  const int wave = (int)((blockIdx.x * blockDim.x + threadIdx.x) >> 5);
  if (wave >= mtiles) return;                    // wave-uniform: EXEC stays all-1
  const int lane = (int)(threadIdx.x & 31);
  const int m    = lane & 15;                    // A row / B,C,D column within tile
  const int half = lane >> 4;
  const bool full_tile = (wave * 16 + 16) <= n_nodes;   // wave-uniform

  // Preload all A fragments for this 16x64 tile of x (16 K-steps of K=4).
  int arow = wave * 16 + m;
  if (arow >= n_nodes) arow = n_nodes - 1;       // clamp (tail-safe); stores guarded
  const float* xrow = x + (size_t)arow * D_FEAT;
  v2f a[16];
#pragma unroll
  for (int ki = 0; ki < 16; ++ki)
    a[ki] = *(const v2f*)(xrow + ki * 4 + half * 2);

#pragma unroll
  for (int w = 0; w < 2; ++w) {
    const float* __restrict__ W = (w == 0) ? W1 : Wr;
    float* __restrict__ dstbuf  = (w == 0) ? h  : out;
#pragma unroll
    for (int nt = 0; nt < 4; ++nt) {
      const int col = nt * 16 + m;               // output column (B column = W row)
      const float* wrow = W + (size_t)col * D_FEAT;   // B[k][col] = W[col][k]
      v8f acc = {};
#pragma unroll
      for (int ki = 0; ki < 16; ++ki) {
        v2f b = *(const v2f*)(wrow + ki * 4 + half * 2);
        acc = __builtin_amdgcn_wmma_f32_16x16x4_f32(
            /*neg_a=*/false, a[ki], /*neg_b=*/false, b,
            /*c_mod=*/(short)0, acc, /*reuse_a=*/false, /*reuse_b=*/false);
      }
      if (w == 1) {
        const float bias = br[col];
#pragma unroll
        for (int r = 0; r < 8; ++r) acc[r] += bias;
      }
      float* dcol = dstbuf + (size_t)(wave * 16 + 8 * half) * D_FEAT + col;
      if (full_tile) {                           // scalar branch, no divergence
#pragma unroll
        for (int r = 0; r < 8; ++r)              // VGPR r -> row (r + 8*half)
          dcol[(size_t)r * D_FEAT] = acc[r];
      } else {
#pragma unroll
        for (int r = 0; r < 8; ++r) {
          const int orow = wave * 16 + r + 8 * half;
          if (orow < n_nodes)
            dstbuf[(size_t)orow * D_FEAT + col] = acc[r];
        }
      }
    }
  }
}

// ---------------------------------------------------------------------------
// Edge scatter: one wave per edge; lane l handles feature pair [2l, 2l+1].
// Coalesced 256B gathers from h, contiguous float atomics into accumulators.
// Lane 0 bumps the two degree counters.
// ---------------------------------------------------------------------------
__global__ __launch_bounds__(256) void relconv_edge_scatter(
    const int* __restrict__ src, const int* __restrict__ dst,
    const float* __restrict__ h,
    float* __restrict__ acc_t, float* __restrict__ acc_s,
    float* __restrict__ cnt_t, float* __restrict__ cnt_s,
    int n_edges) {
  const int e = (int)((blockIdx.x * blockDim.x + threadIdx.x) >> 5);
  if (e >= n_edges) return;
  const int lane = (int)(threadIdx.x & 31);
  const int s = src[e];
  const int t = dst[e];

  const v2f vs = *(const v2f*)(h + (size_t)s * D_FEAT + lane * 2);  // msg s->t
  const v2f vt = *(const v2f*)(h + (size_t)t * D_FEAT + lane * 2);  // msg t->s

  float* at = acc_t + (size_t)t * D_FEAT + lane * 2;
  float* as = acc_s + (size_t)s * D_FEAT + lane * 2;
  atomicAdd(at,     vs.x);
  atomicAdd(at + 1, vs.y);
  atomicAdd(as,     vt.x);
  atomicAdd(as + 1, vt.y);
  if (lane == 0) {
    atomicAdd(cnt_t + t, 1.0f);
    atomicAdd(cnt_s + s, 1.0f);
  }
}

// ---------------------------------------------------------------------------
// Finalize: out += acc_t/max(cnt_t,1) + acc_s/max(cnt_s,1)
// ---------------------------------------------------------------------------
__global__ __launch_bounds__(256) void relconv_finalize(
    const float* __restrict__ acc_t, const float* __restrict__ acc_s,
    const float* __restrict__ cnt_t, const float* __restrict__ cnt_s,
    float* __restrict__ out, int n_elems) {
  const int i = (int)(blockIdx.x * blockDim.x + threadIdx.x);
  if (i >= n_elems) return;
  const int node = i >> 6;                       // D_FEAT == 64
  float ct = cnt_t[node]; ct = (ct < 1.0f) ? 1.0f : ct;
  float cs = cnt_s[node]; cs = (cs < 1.0f) ? 1.0f : cs;
  out[i] += acc_t[i] / ct + acc_s[i] / cs;
}

extern "C" void kernel_launch(void* const* d_in, const int* in_sizes, int n_in,
                              void* d_out, int out_size, void* d_ws, size_t ws_size,
                              hipStream_t stream) {
  const float* x   = (const float*)d_in[0];
  const int*   ei  = (const int*)d_in[1];
  const float* W1  = (const float*)d_in[2];
  const float* Wr  = (const float*)d_in[3];
  const float* br  = (const float*)d_in[4];
  float* out = (float*)d_out;

  const int n_nodes = in_sizes[0] / D_FEAT;
  const int n_edges = in_sizes[1] / 2;
  const int* src = ei;               // edge_index[0]
  const int* dst = ei + n_edges;     // edge_index[1]

  const size_t nfeat = (size_t)n_nodes * D_FEAT;
  char* ws = (char*)d_ws;
  float* h     = (float*)ws;  ws += nfeat * sizeof(float);
  float* acc_t = (float*)ws;  ws += nfeat * sizeof(float);
  float* acc_s = (float*)ws;  ws += nfeat * sizeof(float);
  float* cnt_t = (float*)ws;  ws += (size_t)n_nodes * sizeof(float);
  float* cnt_s = (float*)ws;  ws += (size_t)n_nodes * sizeof(float);

  // Zero accumulators + counters (contiguous region after h). Capture-safe.
  hipMemsetAsync(acc_t, 0, (2 * nfeat + 2 * (size_t)n_nodes) * sizeof(float), stream);

  // GEMMs via WMMA: one wave per 16-row tile, 8 waves / 256-thread block.
  const int mtiles = (n_nodes + 15) / 16;
  const int gblocks = (mtiles + 7) / 8;
  relconv_gemm_wmma<<<gblocks, 256, 0, stream>>>(x, W1, Wr, br, h, out,
                                                 n_nodes, mtiles);

  // Edge scatter: one wave per edge, 8 edges / block.
  const int eblocks = (n_edges + 7) / 8;
  relconv_edge_scatter<<<eblocks, 256, 0, stream>>>(src, dst, h, acc_t, acc_s,
                                                    cnt_t, cnt_s, n_edges);

  // Finalize.
  const int n_elems = n_nodes * D_FEAT;
  relconv_finalize<<<(n_elems + 255) / 256, 256, 0, stream>>>(acc_t, acc_s,
                                                              cnt_t, cnt_s,
                                                              out, n_elems);
}